// GraphSAGEModel_19155554140772
// MI455X (gfx1250) — compile-verified
//
#include <hip/hip_runtime.h>
#include <hip/hip_bf16.h>

typedef __attribute__((ext_vector_type(2))) float v2f;
typedef __attribute__((ext_vector_type(8))) float v8f;

// One fused GraphSAGE layer:
//   out[node] = l2norm(relu(mean_{t<11}(emb[nbr[node][t]]) @ W))
// Block = 256 threads (8 wave32), owns 16 nodes (one 16-row M tile, full DOUT).
// DIN, DOUT multiples of 128; A tile staged in LDS with +4 padding.
template <int DIN, int DOUT>
__global__ __launch_bounds__(256) void sage_layer_kernel(
    const float* __restrict__ emb,   // [Nemb, DIN]
    const int*   __restrict__ nbr,   // [Nnodes, 11]
    const float* __restrict__ W,     // [DIN, DOUT] row-major
    float*       __restrict__ out)   // [Nnodes, DOUT]
{
    constexpr int LDA  = DIN + 4;        // LDS row stride (floats): bank-conflict-free
    constexpr int NSUB = DOUT / 128;     // 16-col WMMA subtiles per wave (8 waves)
    constexpr int CH   = DIN / 64;       // float4 chunks per staging lane

    __shared__ __align__(16) float Atile[16 * LDA];
    __shared__ float rowsq[16];

    const int tid   = threadIdx.x;
    const int node0 = blockIdx.x * 16;

    // ---------- Stage: mean-aggregate 16 node rows into LDS ----------
    {
        const int  g    = tid >> 4;      // node slot 0..15
        const int  l16  = tid & 15;      // column group
        const long node = (long)node0 + g;

        float4 acc[CH];
#pragma unroll
        for (int j = 0; j < CH; ++j) acc[j] = make_float4(0.f, 0.f, 0.f, 0.f);

        for (int t = 0; t < 11; ++t) {
            const long nidx = nbr[node * 11 + t];
            const float4* src = (const float4*)(emb + nidx * (long)DIN);
#pragma unroll
            for (int j = 0; j < CH; ++j) {
                float4 v = src[l16 + 16 * j];
                acc[j].x += v.x; acc[j].y += v.y;
                acc[j].z += v.z; acc[j].w += v.w;
            }
        }
        const float s = 1.0f / 11.0f;
        float4* dst = (float4*)(Atile + g * LDA);   // LDA*4 bytes % 16 == 0
#pragma unroll
        for (int j = 0; j < CH; ++j) {
            float4 v;
            v.x = acc[j].x * s; v.y = acc[j].y * s;
            v.z = acc[j].z * s; v.w = acc[j].w * s;
            dst[l16 + 16 * j] = v;
        }
    }
    if (tid < 16) rowsq[tid] = 0.0f;
    __syncthreads();

    // ---------- GEMM: 16 x DOUT tile, wave w owns cols [w*16*NSUB, ...) ----------
    const int lane = tid & 31;
    const int wave = tid >> 5;
    const int m16  = lane & 15;          // A row / B,C col within subtile
    const int half = lane >> 4;          // 0: rows 0-7 of C, 1: rows 8-15
    const int kk   = half << 1;          // K sub-offset {0,2} within k0..k0+3
    const int colw = wave * (16 * NSUB); // wave's first output column

    v8f c[NSUB];
#pragma unroll
    for (int s = 0; s < NSUB; ++s)
#pragma unroll
        for (int i = 0; i < 8; ++i) c[s][i] = 0.0f;

#pragma unroll 4
    for (int k0 = 0; k0 < DIN; k0 += 4) {
        // A fragment (16x4 f32): lane holds A[m16][k0+kk], A[m16][k0+kk+1]
        const float* ap = Atile + m16 * LDA + k0 + kk;
        v2f a; a.x = ap[0]; a.y = ap[1];
        // B fragments (4x16 f32): lane holds B[k0+kk][n], B[k0+kk+1][n]
        const float* bp = W + (size_t)(k0 + kk) * DOUT + colw + m16;
#pragma unroll
        for (int s = 0; s < NSUB; ++s) {
            v2f b; b.x = bp[s * 16]; b.y = bp[DOUT + s * 16];
            c[s] = __builtin_amdgcn_wmma_f32_16x16x4_f32(
                false, a, false, b, (short)0, c[s], false, false);
        }
    }

    // ---------- Epilogue: relu, row sum-of-squares, l2-normalize ----------
    float ss[8];
#pragma unroll
    for (int v = 0; v < 8; ++v) {
        float t = 0.0f;
#pragma unroll
        for (int s = 0; s < NSUB; ++s) {
            float x = fmaxf(c[s][v], 0.0f);
            c[s][v] = x;
            t += x * x;
        }
        ss[v] = t;
    }
    // reduce over the 16 lanes of each half (cols of this wave)
#pragma unroll
    for (int mk = 1; mk < 16; mk <<= 1) {
#pragma unroll
        for (int v = 0; v < 8; ++v) ss[v] += __shfl_xor(ss[v], mk, 32);
    }
    if (m16 == 0) {
#pragma unroll
        for (int v = 0; v < 8; ++v) atomicAdd(&rowsq[v + 8 * half], ss[v]);
    }
    __syncthreads();

#pragma unroll
    for (int v = 0; v < 8; ++v) {
        const int   row = v + 8 * half;
        const float inv = 1.0f / fmaxf(sqrtf(rowsq[row]), 1e-12f);
        float* op = out + (size_t)(node0 + row) * DOUT + colw + m16;
#pragma unroll
        for (int s = 0; s < NSUB; ++s) op[s * 16] = c[s][v] * inv;
    }
}

extern "C" void kernel_launch(void* const* d_in, const int* in_sizes, int n_in,
                              void* d_out, int out_size, void* d_ws, size_t ws_size,
                              hipStream_t stream) {
    (void)in_sizes; (void)n_in; (void)out_size; (void)ws_size;
    const float* feats = (const float*)d_in[0];   // [262144, 512]
    const int*   nb1   = (const int*)d_in[1];     // [32768, 11]
    const int*   nb2   = (const int*)d_in[2];     // [4096, 11]
    const float* W1    = (const float*)d_in[3];   // [512, 256]
    const float* W2    = (const float*)d_in[4];   // [256, 128]
    float*       out   = (float*)d_out;           // [4096, 128]
    float*       h     = (float*)d_ws;            // [32768, 256] = 32 MB scratch

    constexpr int N2 = 32768, B = 4096;
    sage_layer_kernel<512, 256><<<N2 / 16, 256, 0, stream>>>(feats, nb1, W1, h);
    sage_layer_kernel<256, 128><<<B  / 16, 256, 0, stream>>>(h,     nb2, W2, out);
}